// VectorQuantizer_29454885716150
// MI455X (gfx1250) — compile-verified
//
#include <hip/hip_runtime.h>
#include <hip/hip_bf16.h>

typedef __attribute__((ext_vector_type(2))) float v2f;
typedef __attribute__((ext_vector_type(8))) float v8f;
typedef __attribute__((ext_vector_type(4))) unsigned int u32x4;
typedef __attribute__((ext_vector_type(8))) int i32x8;
typedef __attribute__((ext_vector_type(4))) int i32x4;

#define NPIX   65536      // B*H*W = 16*64*64
#define CDIM   64
#define KCODES 1024
#define HW     4096       // H*W
#define BSTRIDE 262144    // C*H*W
#define LDSP   66         // padded row stride (even -> 8B aligned float2, 66%64=2 -> conflict-free)
#define KT     32         // codes per k-tile (double buffered)
#define NKT    (KCODES / KT)

#if __has_builtin(__builtin_amdgcn_tensor_load_to_lds) && __has_builtin(__builtin_amdgcn_s_wait_tensorcnt)
#define USE_TDM 1
#else
#define USE_TDM 0
#endif

// ---------------- kernel 0: ||e_k||^2 ----------------
__global__ __launch_bounds__(256) void vq_enorm_kernel(const float* __restrict__ emb,
                                                       float* __restrict__ eNorm) {
    int k = blockIdx.x * 256 + threadIdx.x;
    if (k < KCODES) {
        float s = 0.f;
        #pragma unroll
        for (int c = 0; c < CDIM; ++c) {
            float e = emb[k * CDIM + c];
            s = fmaf(e, e, s);
        }
        eNorm[k] = s;
    }
}

#if USE_TDM
// Issue one TDM DMA: 32x64 f32 tile of emb -> LDS, HW-padded to row stride 66 DWORDs
// (pad_interval code 5 = every 64 DWORDs, pad_amount code 1 = 2 DWORDs).
__device__ __forceinline__ void tdm_load_emb_tile(const float* gsrc, unsigned ldsAddr) {
    unsigned long long ga = (unsigned long long)gsrc;
    u32x4 g0;
    g0[0] = 1u;                                                // count=1, user mode
    g0[1] = ldsAddr;                                           // lds_addr [63:32]
    g0[2] = (unsigned)(ga & 0xFFFFFFFFull);                    // global_addr[31:0]
    g0[3] = (unsigned)((ga >> 32) & 0x1FFFFFFull) | (2u << 30);// global_addr[56:32] | type=2
    i32x8 g1;
    g1[0] = (int)((2u << 16)    // data_size = 4B
                | (1u << 20)    // pad_enable
                | (5u << 22)    // pad_interval: 64 DWORDs
                | (1u << 25));  // pad_amount: 2 DWORDs
    g1[1] = (int)(64u << 16);   // tensor_dim0 = 64 (low16 @ bits[63:48])
    g1[2] = (int)(32u << 16);   // tensor_dim0 hi=0; tensor_dim1 = 32 (low16 @ bits[95:80])
    g1[3] = (int)(64u << 16);   // tensor_dim1 hi=0; tile_dim0 = 64 (bits[127:112])
    g1[4] = 32;                 // tile_dim1 = 32; tile_dim2 = 0
    g1[5] = 64;                 // tensor_dim0_stride = 64 (low32)
    g1[6] = 0;                  // stride hi / dim1_stride lo
    g1[7] = 0;                  // dim1_stride hi
    i32x4 gz = {0, 0, 0, 0};
#if defined(__clang_major__) && (__clang_major__ >= 23)
    i32x8 gz8 = {0, 0, 0, 0, 0, 0, 0, 0};
    __builtin_amdgcn_tensor_load_to_lds(g0, g1, gz, gz, gz8, 0);
#else
    __builtin_amdgcn_tensor_load_to_lds(g0, g1, gz, gz, 0);
#endif
}
#endif

// ---------------- kernel 1: WMMA distance + argmin ----------------
// grid = 512 blocks, 256 threads (8 waves). Block handles 128 rows of z_flat.
__global__ __launch_bounds__(256) void vq_argmin_kernel(const float* __restrict__ z,
                                                        const float* __restrict__ emb,
                                                        const float* __restrict__ eNorm,
                                                        int*   __restrict__ nearestI,
                                                        float* __restrict__ nearestF) {
    __shared__ float zTile[128 * LDSP];        // 33,792 B
    __shared__ float embBuf[2][KT * LDSP];     // 2 x 8,448 B (double buffer)

    const int tid   = threadIdx.x;
    const int nBase = blockIdx.x * 128;
    const int b     = nBase >> 12;             // / HW (128 | HW so one image per block)
    const int hw    = nBase & (HW - 1);
    const int zbase = b * BSTRIDE + hw;

    // Transpose-load z tile: coalesced over nl for each channel c.
    #pragma unroll
    for (int i = 0; i < 32; ++i) {
        int lin = tid + i * 256;               // 128*64 = 8192 elements
        int c  = lin >> 7;
        int nl = lin & 127;
        zTile[nl * LDSP + c] = z[zbase + c * HW + nl];
    }

    const int lane = tid & 31;
    const int w    = tid >> 5;                 // wave id 0..7 -> rows w*16..w*16+15
    const int lh   = lane & 15;
    const int half = lane >> 4;                // K'={0,1} vs K'={2,3} of the x4 fragment
    const int aRow = w * 16 + lh;

    float bestV[8];
    int   bestI[8];
    #pragma unroll
    for (int r = 0; r < 8; ++r) { bestV[r] = 3.4e38f; bestI[r] = 0; }

#if USE_TDM
    // Prologue: DMA tile 0 into buffer 0 (single wave issues; TDM ignores EXEC).
    if (w == 0) {
        tdm_load_emb_tile(emb, (unsigned)(unsigned long long)(void*)&embBuf[0][0]);
    }
#endif

    for (int kt = 0; kt < NKT; ++kt) {
        const int kbase = kt * KT;
        float* eb = &embBuf[kt & 1][0];
        __syncthreads();   // all readers of buf[(kt+1)&1] (prev iter) done; zTile ready at kt=0
#if USE_TDM
        if (w == 0) {
            if (kt + 1 < NKT) {
                tdm_load_emb_tile(emb + (kbase + KT) * CDIM,
                                  (unsigned)(unsigned long long)(void*)&embBuf[(kt + 1) & 1][0]);
                __builtin_amdgcn_s_wait_tensorcnt(1);  // in-order TDM: tile kt complete
            } else {
                __builtin_amdgcn_s_wait_tensorcnt(0);
            }
        }
#else
        #pragma unroll
        for (int i = 0; i < 8; ++i) {
            int lin = tid + i * 256;           // 32*64 = 2048 elements
            int kl = lin >> 6;
            int c  = lin & 63;
            eb[kl * LDSP + c] = emb[(kbase + kl) * CDIM + c];
        }
#endif
        __syncthreads();                       // publish tile kt to all waves

        #pragma unroll
        for (int t = 0; t < 2; ++t) {          // 2 column sub-tiles of 16 codes
            v8f acc = {0.f, 0.f, 0.f, 0.f, 0.f, 0.f, 0.f, 0.f};
            #pragma unroll
            for (int cc = 0; cc < 16; ++cc) {  // C=64 in K'=4 chunks
                // A 16x4 f32: lanes 0-15 K'={0,1}, lanes 16-31 K'={2,3}
                v2f a  = *(const v2f*)&zTile[aRow * LDSP + (cc << 2) + (half << 1)];
                // B 4x16 f32: element (k', n) = emb[code n][c]; mirrored layout
                v2f bb = *(const v2f*)&eb[(t * 16 + lh) * LDSP + (cc << 2) + (half << 1)];
                acc = __builtin_amdgcn_wmma_f32_16x16x4_f32(
                        false, a, false, bb, (short)0, acc, false, false);
            }
            const int   kcol = kbase + t * 16 + lh;   // this lane's code column
            const float en   = eNorm[kcol];
            #pragma unroll
            for (int r = 0; r < 8; ++r) {
                // score = ||e||^2 - 2*dot  (row-constant ||z||^2 dropped: argmin-invariant)
                float s = fmaf(-2.0f, acc[r], en);
                if (s < bestV[r]) { bestV[r] = s; bestI[r] = kcol; }
            }
        }
    }

    // Min-reduce across the 16 lanes holding each row (xor<16 stays within half).
    #pragma unroll
    for (int r = 0; r < 8; ++r) {
        float v  = bestV[r];
        int   ix = bestI[r];
        #pragma unroll
        for (int m = 8; m >= 1; m >>= 1) {
            float ov = __shfl_xor(v, m, 32);
            int   oi = __shfl_xor(ix, m, 32);
            if (ov < v || (ov == v && oi < ix)) { v = ov; ix = oi; }
        }
        if (lh == 0) {
            // C/D layout: VGPR r -> row r (lanes 0-15) / row r+8 (lanes 16-31)
            int row = nBase + w * 16 + r + half * 8;
            nearestI[row] = ix;
            nearestF[row] = (float)ix;
        }
    }
}

// ---------------- kernel 2: gather z_q, write out / z_flat, partial loss ----------------
// grid = 1024 blocks, 256 threads. Block handles 64 rows.
__global__ __launch_bounds__(256) void vq_gather_kernel(const float* __restrict__ z,
                                                        const float* __restrict__ emb,
                                                        const int*   __restrict__ nearestI,
                                                        float* __restrict__ outT,
                                                        float* __restrict__ zflat,
                                                        float* __restrict__ partials) {
    __shared__ float zTile[64 * LDSP];
    __shared__ float zqTile[64 * LDSP];
    __shared__ float wsum[8];

    const int tid   = threadIdx.x;
    const int nBase = blockIdx.x * 64;
    const int b     = nBase >> 12;
    const int hw    = nBase & (HW - 1);
    const int zbase = b * BSTRIDE + hw;

    // Load z tile (coalesced over nl per channel).
    #pragma unroll
    for (int i = 0; i < 16; ++i) {
        int lin = tid + i * 256;          // 64*64 = 4096
        int c  = lin >> 6;
        int nl = lin & 63;
        zTile[nl * LDSP + c] = z[zbase + c * HW + nl];
    }
    __syncthreads();

    float acc = 0.f;
    // nl-major pass: gather emb rows (coalesced over c), emit z_flat, accumulate MSE.
    #pragma unroll
    for (int i = 0; i < 16; ++i) {
        int lin = tid + i * 256;
        int nl = lin >> 6;
        int c  = lin & 63;
        int k  = nearestI[nBase + nl];    // broadcast within wave
        float q  = emb[k * CDIM + c];
        float zv = zTile[nl * LDSP + c];
        zqTile[nl * LDSP + c] = q;
        zflat[(nBase + nl) * CDIM + c] = zv;
        float d = q - zv;
        acc = fmaf(d, d, acc);
    }
    __syncthreads();

    // c-major pass: write NCHW output (coalesced over nl).
    #pragma unroll
    for (int i = 0; i < 16; ++i) {
        int lin = tid + i * 256;
        int c  = lin >> 6;
        int nl = lin & 63;
        outT[zbase + c * HW + nl] = zqTile[nl * LDSP + c];
    }

    // block reduction of squared-diff partial sum
    #pragma unroll
    for (int m = 16; m >= 1; m >>= 1) acc += __shfl_xor(acc, m, 32);
    if ((tid & 31) == 0) wsum[tid >> 5] = acc;
    __syncthreads();
    if (tid == 0) {
        float s = 0.f;
        #pragma unroll
        for (int j = 0; j < 8; ++j) s += wsum[j];
        partials[blockIdx.x] = s;
    }
}

// ---------------- kernel 3: deterministic loss reduction ----------------
__global__ __launch_bounds__(256) void vq_loss_kernel(const float* __restrict__ partials,
                                                      float* __restrict__ lossOut,
                                                      int nPartials) {
    __shared__ float sh[256];
    int tid = threadIdx.x;
    float s = 0.f;
    for (int i = tid; i < nPartials; i += 256) s += partials[i];
    sh[tid] = s;
    __syncthreads();
    for (int off = 128; off > 0; off >>= 1) {
        if (tid < off) sh[tid] += sh[tid + off];
        __syncthreads();
    }
    if (tid == 0) {
        // loss = mean((zq-z)^2) + 0.25*mean((zq-z)^2) = 1.25 * sum / (N*C)
        lossOut[0] = sh[0] * (1.25f / (float)(NPIX * CDIM));
    }
}

extern "C" void kernel_launch(void* const* d_in, const int* in_sizes, int n_in,
                              void* d_out, int out_size, void* d_ws, size_t ws_size,
                              hipStream_t stream) {
    const float* z   = (const float*)d_in[0];   // [16,64,64,64]
    const float* emb = (const float*)d_in[1];   // [1024,64]

    float* out      = (float*)d_out;            // [B,C,H,W] = 4,194,304
    float* loss     = out + 4194304;            // scalar
    float* nearestF = out + 4194305;            // 65,536 (argmin indices as float)
    float* zflat    = out + 4194305 + NPIX;     // 4,194,304

    float* eNorm    = (float*)d_ws;                               // 1024 f
    int*   nearestI = (int*)((char*)d_ws + 4096);                 // 65536 i32
    float* partials = (float*)((char*)d_ws + 4096 + NPIX * 4);    // 1024 f

    vq_enorm_kernel<<<(KCODES + 255) / 256, 256, 0, stream>>>(emb, eNorm);
    vq_argmin_kernel<<<NPIX / 128, 256, 0, stream>>>(z, emb, eNorm, nearestI, nearestF);
    vq_gather_kernel<<<NPIX / 64, 256, 0, stream>>>(z, emb, nearestI, out, zflat, partials);
    vq_loss_kernel<<<1, 256, 0, stream>>>(partials, loss, NPIX / 64);
}